// Deepeucloss_21010980012213
// MI455X (gfx1250) — compile-verified
//
#include <hip/hip_runtime.h>
#include <stdint.h>

// ---------------------------------------------------------------------------
// Deep euclidean + KL loss reduction for MI455X (gfx1250).
// Pure streaming reduction: 5 x 25.17M f32 ~ 503 MB read once -> ~21.6 us HBM
// floor @ 23.3 TB/s. Bandwidth-bound; uses CDNA5 async global->LDS copies
// (ASYNCcnt) with double buffering when the builtins are available.
// ---------------------------------------------------------------------------

typedef float v4f __attribute__((ext_vector_type(4)));
typedef int v4i __attribute__((ext_vector_type(4)));

#define ROWS    2048          // 16 batches * 128 points
#define ROWLEN  12288         // 4096 * 3 elements per row
#define NTH     256           // threads per block (8 waves of 32)
#define CHUNKS  12            // ROWLEN / (NTH * 4)

#if defined(__HIP_DEVICE_COMPILE__) && \
    __has_builtin(__builtin_amdgcn_global_load_async_to_lds_b128)
#define HAVE_ASYNC 1
#else
#define HAVE_ASYNC 0
#endif

#if HAVE_ASYNC
// Builtin signature (from compiler diagnostic): typed v4i pointers in the
// global (AS1) and LDS (AS3) address spaces, plus imm offset and imm cpol.
typedef __attribute__((address_space(1))) v4i as1_v4i;
typedef __attribute__((address_space(3))) v4i as3_v4i;

__device__ __forceinline__ as1_v4i* as_glob(const void* p) {
  // Generic (flat) global pointer == virtual address; inttoptr into AS1.
  return (as1_v4i*)(uintptr_t)p;
}
__device__ __forceinline__ as3_v4i* as_lds(const void* p) {
  // Flat LDS pointer: low 32 bits are the wave-relative LDS byte offset.
  return (as3_v4i*)(uint32_t)(uintptr_t)p;
}

#if __has_builtin(__builtin_amdgcn_s_wait_asynccnt)
#define WAIT_ASYNC(n) __builtin_amdgcn_s_wait_asynccnt(n)
#else
#define WAIT_ASYNC(n) asm volatile("s_wait_asynccnt %0" ::"i"(n) : "memory")
#endif
#endif // HAVE_ASYNC

// Per-element accumulation: squared errors for 3 prediction streams plus the
// closed-form KL base term for the variance stream.
__device__ __forceinline__ void accum_elems(const v4f o, const v4f tg,
                                            const v4f a1, const v4f a2,
                                            const v4f vv, float& so, float& s1,
                                            float& s2, float& sb) {
#pragma unroll
  for (int k = 0; k < 4; ++k) {
    float d0 = o[k] - tg[k];
    float d1 = a1[k] - tg[k];
    float d2 = a2[k] - tg[k];
    so = fmaf(d0, d0, so);
    s1 = fmaf(d1, d1, s1);
    s2 = fmaf(d2, d2, s2);
    float v = vv[k];
    // log(2/v) + v^2/8 - 0.5  ==  (ln2 - 0.5) - ln(v) + 0.125*v^2
    sb += 0.19314718055994531f - logf(v) + v * v * 0.125f;
  }
}

__global__ __launch_bounds__(NTH) void row_reduce_kernel(
    const float* __restrict__ out, const float* __restrict__ target,
    const float* __restrict__ g1, const float* __restrict__ g2,
    const float* __restrict__ var, float* __restrict__ ws) {
  const int t = threadIdx.x;
  const int row = blockIdx.x;
  const size_t base = (size_t)row * ROWLEN;

  float so = 0.f, s1 = 0.f, s2 = 0.f, sb = 0.f;

#if HAVE_ASYNC
  // Double-buffered async global->LDS streaming: 2 buffers x 5 arrays x
  // 256 lanes x 16B = 40 KB LDS. Each lane only reads the LDS it requested,
  // so s_wait_asynccnt is the only synchronization needed.
  __shared__ v4f buf[2][5][NTH];
  const float* gp[5] = {out + base, target + base, g1 + base, g2 + base,
                        var + base};
  {
    const int off = t * 4;
#pragma unroll
    for (int a = 0; a < 5; ++a)
      __builtin_amdgcn_global_load_async_to_lds_b128(
          as_glob(gp[a] + off), as_lds(&buf[0][a][t]), 0, 0);
  }
  for (int c = 0; c < CHUNKS; ++c) {
    if (c + 1 < CHUNKS) {
      const int off = ((c + 1) * NTH + t) * 4;
#pragma unroll
      for (int a = 0; a < 5; ++a)
        __builtin_amdgcn_global_load_async_to_lds_b128(
            as_glob(gp[a] + off), as_lds(&buf[(c + 1) & 1][a][t]), 0, 0);
      WAIT_ASYNC(5);  // 5 newest (next chunk) may remain in flight
    } else {
      WAIT_ASYNC(0);
    }
    asm volatile("" ::: "memory");
    const int pb = c & 1;
    v4f o = buf[pb][0][t];
    v4f tg = buf[pb][1][t];
    v4f a1 = buf[pb][2][t];
    v4f a2 = buf[pb][3][t];
    v4f vv = buf[pb][4][t];
    accum_elems(o, tg, a1, a2, vv, so, s1, s2, sb);
  }
#else
  // Fallback: direct non-temporal b128 streaming loads.
  for (int c = 0; c < CHUNKS; ++c) {
    const size_t off = base + ((size_t)c * NTH + t) * 4;
    v4f o = __builtin_nontemporal_load((const v4f*)(out + off));
    v4f tg = __builtin_nontemporal_load((const v4f*)(target + off));
    v4f a1 = __builtin_nontemporal_load((const v4f*)(g1 + off));
    v4f a2 = __builtin_nontemporal_load((const v4f*)(g2 + off));
    v4f vv = __builtin_nontemporal_load((const v4f*)(var + off));
    accum_elems(o, tg, a1, a2, vv, so, s1, s2, sb);
  }
#endif

  // Block tree reduction of the four row statistics.
  __shared__ float red[4][NTH];
  red[0][t] = so;
  red[1][t] = s1;
  red[2][t] = s2;
  red[3][t] = sb;
  __syncthreads();
  for (int s = NTH / 2; s > 0; s >>= 1) {
    if (t < s) {
      red[0][t] += red[0][t + s];
      red[1][t] += red[1][t + s];
      red[2][t] += red[2][t + s];
      red[3][t] += red[3][t + s];
    }
    __syncthreads();
  }
  if (t == 0) {
    v4f r = {red[0][0], red[1][0], red[2][0], red[3][0]};
    ((v4f*)ws)[row] = r;
  }
}

__global__ __launch_bounds__(NTH) void final_reduce_kernel(
    const float* __restrict__ ws, const float* __restrict__ gt0p,
    const float* __restrict__ legp, const float* __restrict__ ldynp,
    float* __restrict__ outp) {
  const int t = threadIdx.x;
  float sO = 0.f, s1 = 0.f, s2 = 0.f, sB = 0.f;  // raw sums (KL numerators)
  float qO = 0.f, q1 = 0.f, q2 = 0.f;            // sums of per-row sqrt (euc)
  for (int r = t; r < ROWS; r += NTH) {
    v4f v = ((const v4f*)ws)[r];
    sO += v.x;
    s1 += v.y;
    s2 += v.z;
    sB += v.w;
    qO += sqrtf(v.x);
    q1 += sqrtf(v.y);
    q2 += sqrtf(v.z);
  }
  __shared__ float red[7][NTH];
  red[0][t] = sO;
  red[1][t] = s1;
  red[2][t] = s2;
  red[3][t] = sB;
  red[4][t] = qO;
  red[5][t] = q1;
  red[6][t] = q2;
  __syncthreads();
  for (int s = NTH / 2; s > 0; s >>= 1) {
    if (t < s) {
#pragma unroll
      for (int j = 0; j < 7; ++j) red[j][t] += red[j][t + s];
    }
    __syncthreads();
  }
  if (t == 0) {
    const float inv_nc = 1.0f / 128.0f;  // 1 / NUM_CLASSES
    float eucO = red[4][0] * inv_nc;
    float euc1 = red[5][0] * inv_nc;
    float euc2 = red[6][0] * inv_nc;
    float gt_loss = 0.1f * euc1 + 0.2f * euc2;
    // kl = 1.4 * sum(base) + (sumSq_out + 0.2*(sumSq_g1 + sumSq_g2)) / 8
    float kl = 1.4f * red[3][0] +
               (red[0][0] + 0.2f * (red[1][0] + red[2][0])) * 0.125f;
    float l_dyn = *ldynp;
    float leg_term = 0.01f * 0.2f * l_dyn * (*legp);  // stop-grad constant
    float reg = (*gt0p) * 0.01f * l_dyn;              // stop-grad constant
    float res =
        eucO + leg_term + gt_loss + reg + kl / (1.2f * (eucO + gt_loss));
    *outp = res;
  }
}

extern "C" void kernel_launch(void* const* d_in, const int* in_sizes, int n_in,
                              void* d_out, int out_size, void* d_ws,
                              size_t ws_size, hipStream_t stream) {
  (void)in_sizes;
  (void)n_in;
  (void)out_size;
  (void)ws_size;
  const float* out_p = (const float*)d_in[0];
  const float* target_p = (const float*)d_in[1];
  const float* gt0_p = (const float*)d_in[2];
  const float* g1_p = (const float*)d_in[3];
  const float* g2_p = (const float*)d_in[4];
  const float* var_p = (const float*)d_in[5];
  const float* leg_p = (const float*)d_in[6];
  const float* ldyn_p = (const float*)d_in[7];
  float* ws = (float*)d_ws;  // ROWS * 4 floats = 32 KB

  row_reduce_kernel<<<ROWS, NTH, 0, stream>>>(out_p, target_p, g1_p, g2_p,
                                              var_p, ws);
  final_reduce_kernel<<<1, NTH, 0, stream>>>(ws, gt0_p, leg_p, ldyn_p,
                                             (float*)d_out);
}